// GCN_74045236183290
// MI455X (gfx1250) — compile-verified
//
#include <hip/hip_runtime.h>

typedef __attribute__((ext_vector_type(2))) float v2f;
typedef __attribute__((ext_vector_type(8))) float v8f;

#define NNODES 50000
#define NEDGES 800000
#define DIM 64

// ---------------- degree / normalization ----------------

__global__ __launch_bounds__(256) void deg_init_kernel(float* deg, int n) {
    int i = blockIdx.x * 256 + threadIdx.x;
    if (i < n) deg[i] = 1.0f;  // self loop contributes 1 to deg(row)
}

__global__ __launch_bounds__(256) void deg_count_kernel(const long long* __restrict__ rows,
                                                        float* deg, int e) {
    int i = blockIdx.x * 256 + threadIdx.x;
    if (i < e) atomicAdd(&deg[(int)rows[i]], 1.0f);
}

__global__ __launch_bounds__(256) void deg_rsqrt_kernel(float* deg, int n) {
    int i = blockIdx.x * 256 + threadIdx.x;
    if (i < n) deg[i] = rsqrtf(deg[i]);  // deg >= 1 always (self loops)
}

// ---------------- h = x @ W via V_WMMA_F32_16X16X4_F32 ----------------
// Each block: 256 threads = 8 waves. Weight (64x64 f32, 16KB) staged in LDS.
// Each wave computes a 16-row M-tile across all 64 output columns using
// 4 f32 accumulators (4 x 16 columns), stepping K by 4 per WMMA.
//
// A (16x4 f32) lane layout: lanes 0-15 hold row M=lane, K=(kk+0,kk+1) in
// (v0,v1); lanes 16-31 hold row M=lane-16, K=(kk+2,kk+3).
// B (4x16 f32) mirrors: lanes 0-15 hold col N=lane, K=(kk+0,kk+1);
// lanes 16-31 hold col N=lane-16, K=(kk+2,kk+3).
// D (16x16 f32): VGPR r, lanes 0-15 -> (M=r, N=lane); lanes 16-31 -> (M=8+r).

__global__ __launch_bounds__(256) void gemm_wmma_f32_kernel(const float* __restrict__ x,
                                                            const float* __restrict__ w,
                                                            float* __restrict__ h,
                                                            int n_nodes) {
    __shared__ float wlds[DIM * DIM];
    const int tid = threadIdx.x;

    // cooperative 16KB weight load into LDS (float4-vectorized)
    const float4* wsrc = (const float4*)w;
    float4* wdst = (float4*)wlds;
    #pragma unroll
    for (int i = tid; i < (DIM * DIM) / 4; i += 256) wdst[i] = wsrc[i];
    __syncthreads();

    const int wave  = tid >> 5;
    const int lane  = tid & 31;
    const int lhalf = lane & 15;
    const int khi   = lane >> 4;            // 0: lanes 0-15, 1: lanes 16-31

    const int m0 = (blockIdx.x * 8 + wave) * 16;
    if (m0 >= n_nodes) return;

    v8f acc0 = {}, acc1 = {}, acc2 = {}, acc3 = {};

    const float* xrow = x + (size_t)(m0 + lhalf) * DIM;

    for (int kk = 0; kk < DIM; kk += 4) {
        const int ka = kk + 2 * khi;
        v2f a;
        a.x = xrow[ka + 0];
        a.y = xrow[ka + 1];

        v2f b0, b1, b2, b3;
        b0.x = wlds[(ka + 0) * DIM + 0  + lhalf];
        b0.y = wlds[(ka + 1) * DIM + 0  + lhalf];
        b1.x = wlds[(ka + 0) * DIM + 16 + lhalf];
        b1.y = wlds[(ka + 1) * DIM + 16 + lhalf];
        b2.x = wlds[(ka + 0) * DIM + 32 + lhalf];
        b2.y = wlds[(ka + 1) * DIM + 32 + lhalf];
        b3.x = wlds[(ka + 0) * DIM + 48 + lhalf];
        b3.y = wlds[(ka + 1) * DIM + 48 + lhalf];

        acc0 = __builtin_amdgcn_wmma_f32_16x16x4_f32(false, a, false, b0, (short)0, acc0, false, false);
        acc1 = __builtin_amdgcn_wmma_f32_16x16x4_f32(false, a, false, b1, (short)0, acc1, false, false);
        acc2 = __builtin_amdgcn_wmma_f32_16x16x4_f32(false, a, false, b2, (short)0, acc2, false, false);
        acc3 = __builtin_amdgcn_wmma_f32_16x16x4_f32(false, a, false, b3, (short)0, acc3, false, false);
    }

    const int mbase = m0 + khi * 8;
    float* hrow = h + (size_t)mbase * DIM + lhalf;
    #pragma unroll
    for (int r = 0; r < 8; ++r) {
        hrow[(size_t)r * DIM + 0]  = acc0[r];
        hrow[(size_t)r * DIM + 16] = acc1[r];
        hrow[(size_t)r * DIM + 32] = acc2[r];
        hrow[(size_t)r * DIM + 48] = acc3[r];
    }
}

// ---------------- out init: bias + self-loop term ----------------

__global__ __launch_bounds__(256) void out_init_kernel(const float* __restrict__ h,
                                                       const float* __restrict__ dinv,
                                                       const float* __restrict__ bias,
                                                       float* __restrict__ out, int n) {
    int t = blockIdx.x * 256 + threadIdx.x;
    if (t >= n * DIM) return;
    int i = t >> 6;
    int d = t & 63;
    float di = dinv[i];
    out[t] = bias[d] + h[t] * di * di;
}

// ---------------- edge scatter: out[col] += h[row] * norm ----------------
// 64 threads per edge (element-per-thread): coalesced 256B gather of h[row],
// global_atomic_add_f32 scatter into out[col].

__global__ __launch_bounds__(256) void scatter_kernel(const long long* __restrict__ rows,
                                                      const long long* __restrict__ cols,
                                                      const float* __restrict__ h,
                                                      const float* __restrict__ dinv,
                                                      float* __restrict__ out, int e) {
    int t = blockIdx.x * 256 + threadIdx.x;
    int edge = t >> 6;
    if (edge >= e) return;
    int d = t & 63;
    int r = (int)rows[edge];
    int c = (int)cols[edge];
    float nrm = dinv[r] * dinv[c];
    atomicAdd(&out[(size_t)c * DIM + d], h[(size_t)r * DIM + d] * nrm);
}

// ---------------- launcher ----------------

extern "C" void kernel_launch(void* const* d_in, const int* in_sizes, int n_in,
                              void* d_out, int out_size, void* d_ws, size_t ws_size,
                              hipStream_t stream) {
    const float*     x    = (const float*)d_in[0];
    const long long* ei   = (const long long*)d_in[1];  // int64 [2, E]
    const float*     w    = (const float*)d_in[2];
    const float*     bias = (const float*)d_in[3];
    float*           out  = (float*)d_out;

    const long long* rows = ei;           // edge_index[0]
    const long long* cols = ei + NEDGES;  // edge_index[1]

    float* dinv = (float*)d_ws;                 // [NNODES], deg then rsqrt in place
    float* h    = dinv + NNODES;                // [NNODES * DIM]

    deg_init_kernel<<<(NNODES + 255) / 256, 256, 0, stream>>>(dinv, NNODES);
    deg_count_kernel<<<(NEDGES + 255) / 256, 256, 0, stream>>>(rows, dinv, NEDGES);
    deg_rsqrt_kernel<<<(NNODES + 255) / 256, 256, 0, stream>>>(dinv, NNODES);

    const int mtiles = NNODES / 16;  // 3125, exact
    gemm_wmma_f32_kernel<<<(mtiles + 7) / 8, 256, 0, stream>>>(x, w, h, NNODES);

    out_init_kernel<<<(NNODES * DIM + 255) / 256, 256, 0, stream>>>(h, dinv, bias, out, NNODES);
    scatter_kernel<<<((size_t)NEDGES * DIM + 255) / 256, 256, 0, stream>>>(rows, cols, h, dinv, out, NEDGES);
}